// LimbLength_53008486367482
// MI455X (gfx1250) — compile-verified
//
#include <hip/hip_runtime.h>

// LimbLength: out[b,j] = || in[b,j,:] - in[b,parent[j],:] ||_2
// B=500000, K=15, D=3, fp32. Pure bandwidth kernel (~120 MB @ 23.3 TB/s ~ 5.2us).
// Strategy: stage 256-body tiles (46080 B) through LDS with CDNA5 async
// global->LDS b128 DMA (ASYNCcnt), compute from LDS (conflict-free: stride 45
// dwords is odd -> bijective over 64 banks), write results to LDS and scatter
// back with async LDS->global b128 DMA.

#define TPB      256
#define BODIES   256
#define IN_F     45                      // 15 joints * 3 coords
#define OUT_F    15
#define IN_VEC4  ((BODIES * IN_F) / 4)   // 2880 float4 per tile (16B-aligned chunk)
#define OUT_VEC4 ((BODIES * OUT_F) / 4)  // 960  float4 per tile

// Builtin parameter types (from hipcc diagnostic): pointer-to-int4 with
// explicit address spaces: AS1 (global, printed "__device__") and AS3 (LDS).
typedef int v4i __attribute__((ext_vector_type(4)));
typedef __attribute__((address_space(1))) v4i* gbl_v4i_p;
typedef __attribute__((address_space(3))) v4i* lds_v4i_p;

// Flat LDS address = {shared_aperture[63:32], offset[31:0]} (ISA 10.2): the low
// 32 bits of a generic pointer to __shared__ data are the LDS byte address.
static __device__ __forceinline__ unsigned lds_off(const void* p) {
  return (unsigned)(unsigned long long)p;
}
static __device__ __forceinline__ lds_v4i_p as_lds(const void* p) {
  return (lds_v4i_p)(unsigned long long)lds_off(p);
}
// For global memory, flat address == global address numerically.
static __device__ __forceinline__ gbl_v4i_p as_gbl(const void* p) {
  return (gbl_v4i_p)(unsigned long long)p;
}

static __device__ __forceinline__ void async_load16(const float4* g, float4* l) {
#if __has_builtin(__builtin_amdgcn_global_load_async_to_lds_b128)
  __builtin_amdgcn_global_load_async_to_lds_b128(as_gbl(g), as_lds(l), 0, 0);
#else
  asm volatile("global_load_async_to_lds_b128 %0, %1, off"
               :: "v"(lds_off(l)), "v"(g) : "memory");
#endif
}

static __device__ __forceinline__ void async_store16(float4* g, const float4* l) {
#if __has_builtin(__builtin_amdgcn_global_store_async_from_lds_b128)
  __builtin_amdgcn_global_store_async_from_lds_b128(as_gbl(g), as_lds(l), 0, 0);
#else
  asm volatile("global_store_async_from_lds_b128 %0, %1, off"
               :: "v"(g), "v"(lds_off(l)) : "memory");
#endif
}

static __device__ __forceinline__ void wait_async0() {
#if __has_builtin(__builtin_amdgcn_s_wait_asynccnt)
  __builtin_amdgcn_s_wait_asynccnt(0);
#else
  asm volatile("s_wait_asynccnt 0" ::: "memory");
#endif
}

__global__ __launch_bounds__(TPB)
void LimbLength_kernel(const float* __restrict__ in, float* __restrict__ out, int B) {
  __shared__ __align__(16) float s_in[BODIES * IN_F];    // 46080 B
  __shared__ __align__(16) float s_out[BODIES * OUT_F];  // 15360 B

  const int tid = threadIdx.x;
  const long long body0 = (long long)blockIdx.x * BODIES;
  const long long rem   = (long long)B - body0;
  const int nb = (rem < (long long)BODIES) ? (int)rem : BODIES;

  // ---- Gather tile into LDS ----
  if (nb == BODIES) {
    // Tile base = blk * 46080 bytes -> 16B aligned; lane-consecutive float4s
    // give fully-coalesced 512B bursts per async b128 DMA instruction.
    const float4* gsrc = (const float4*)(in + body0 * IN_F);
    float4* lin = (float4*)s_in;
    for (int i = tid; i < IN_VEC4; i += TPB) async_load16(gsrc + i, lin + i);
    wait_async0();
  } else {
    // Ragged tail block: plain coalesced dword loads.
    const int n = nb * IN_F;
    const float* gsrc = in + body0 * IN_F;
    for (int i = tid; i < n; i += TPB) s_in[i] = gsrc[i];
  }
  __syncthreads();

  // ---- Compute: one thread per body ----
  if (tid < nb) {
    float v[IN_F];
    const float* p = &s_in[tid * IN_F];  // stride 45 dwords: conflict-free banks
#pragma unroll
    for (int i = 0; i < IN_F; ++i) v[i] = p[i];

    const int par[OUT_F] = {0, 0, 1, 1, 1, 3, 4, 5, 6, 2, 2, 9, 10, 11, 12};
    float* o = &s_out[tid * OUT_F];
#pragma unroll
    for (int j = 0; j < OUT_F; ++j) {
      const int pj = par[j];
      const float dx = v[3 * j + 0] - v[3 * pj + 0];
      const float dy = v[3 * j + 1] - v[3 * pj + 1];
      const float dz = v[3 * j + 2] - v[3 * pj + 2];
      o[j] = __builtin_sqrtf(dx * dx + dy * dy + dz * dz);
    }
  }
  __syncthreads();

  // ---- Scatter tile back ----
  if (nb == BODIES) {
    float4* gdst = (float4*)(out + body0 * OUT_F);  // blk*15360 B -> 16B aligned
    const float4* lo = (const float4*)s_out;
    for (int i = tid; i < OUT_VEC4; i += TPB) async_store16(gdst + i, lo + i);
    // S_ENDPGM performs an implicit wait-idle: ASYNCcnt drains before the
    // workgroup's LDS is released (CDNA5 ISA).
  } else {
    const int n = nb * OUT_F;
    float* gdst = out + body0 * OUT_F;
    for (int i = tid; i < n; i += TPB) gdst[i] = s_out[i];
  }
}

extern "C" void kernel_launch(void* const* d_in, const int* in_sizes, int n_in,
                              void* d_out, int out_size, void* d_ws, size_t ws_size,
                              hipStream_t stream) {
  (void)n_in; (void)out_size; (void)d_ws; (void)ws_size;
  const float* in = (const float*)d_in[0];
  float* out = (float*)d_out;
  const int B = in_sizes[0] / IN_F;  // 22,500,000 / 45 = 500,000 bodies
  const int blocks = (B + BODIES - 1) / BODIES;
  LimbLength_kernel<<<blocks, TPB, 0, stream>>>(in, out, B);
}